// DUC_89206470738669
// MI455X (gfx1250) — compile-verified
//
#include <hip/hip_runtime.h>
#include <cstdint>
#include <cstddef>

// DUC pixel-shuffle (r=8) + zero-pad to 304x304.
//   in : [16][19*64][32][32] f32   (bc = b*19 + c, channel block = bc*64)
//   out: [16][19][304][304]  f32
//
// Pure data movement (0 FLOPs): ~76 MiB read + ~107 MiB write -> ~8 us at
// 23.3 TB/s HBM. Split into:
//  1) interior kernel: power-of-two index space (bc<<14 | oh<<6 | w4) so all
//     coordinate math is shifts/masks; 4x b32 gathers at 4 KiB channel stride
//     (absorbed by the 192 MB L2) + coalesced 128-bit non-temporal stores.
//  2) pad kernel: streams ~31 MiB of zeros (right strips + bottom rows) with
//     NT b128 stores.

namespace {
constexpr int kBC        = 16 * 19;   // 304 (b,c) pairs
constexpr int kOut       = 304;       // padded canvas edge
constexpr int kInterior  = 256;       // 32 * 8
constexpr int kW4Out     = kOut / 4;        // 76 float4 per output row
constexpr int kW4In      = kInterior / 4;   // 64 interior float4 per row
constexpr int kRow4      = kOut * kW4Out;   // 23104 float4 per (b,c) image
constexpr int kBlock     = 256;             // 8 wave32 waves

// Interior: bc * 256 rows * 64 float4  (indexed bc<<14 | oh<<6 | w4)
constexpr int kInterior4 = kBC * kInterior * kW4In;      // 4,980,736
// Pad: per bc -> 256 rows * 12 float4 (right) + 48 rows * 76 float4 (bottom)
constexpr int kPadRight  = kInterior * (kW4Out - kW4In); // 3072
constexpr int kPadPerBC  = kPadRight + (kOut - kInterior) * kW4Out; // 6720
constexpr int kPad4      = kBC * kPadPerBC;              // 2,042,880

static_assert(kInterior4 % kBlock == 0, "interior grid exact");
static_assert(kPad4 % kBlock == 0, "pad grid exact");
}  // namespace

typedef float v4f __attribute__((ext_vector_type(4)));

__global__ __launch_bounds__(kBlock) void duc_interior_kernel(
    const float* __restrict__ x, float* __restrict__ out) {
  const int idx = blockIdx.x * kBlock + threadIdx.x;
  const int w4 = idx & 63;            // interior float4 within row
  const int oh = (idx >> 6) & 255;    // output row
  const int bc = idx >> 14;           // (b,c) pair

  // Input dword offset as a pure bitfield merge:
  //   ch = bc*64 + (oh&7)*8 + (w4&1)*4 ; i = oh>>3 ; j = w4>>1
  //   off = ch*1024 + i*32 + j
  const int in_off = (bc << 16) | ((oh & 7) << 13) | ((w4 & 1) << 12) |
                     ((oh >> 3) << 5) | (w4 >> 1);
  const float* __restrict__ p = x + in_off;

  v4f v;
  v.x = p[0 * 1024];   // 4 gathers, consecutive channels (4 KiB stride);
  v.y = p[1 * 1024];   // consecutive lanes read consecutive j -> each 128B
  v.z = p[2 * 1024];   // input line is consumed densely out of L2.
  v.w = p[3 * 1024];

  // gfx1250 global_prefetch_b8: next i-row line, consumed by blocks at oh+8.
  __builtin_prefetch(p + 32, 0, 3);

  // Coalesced 128-bit streaming store (th:NT) — output is write-once.
  const int out4 = bc * kRow4 + oh * kW4Out + w4;
  __builtin_nontemporal_store(v, reinterpret_cast<v4f*>(out) + out4);
}

__global__ __launch_bounds__(kBlock) void duc_pad_kernel(
    float* __restrict__ out) {
  const int idx = blockIdx.x * kBlock + threadIdx.x;
  const int bc = idx / kPadPerBC;
  const int p  = idx % kPadPerBC;

  int out4;
  if (p < kPadRight) {
    // Right strip: rows 0..255, float4 columns 64..75
    const int oh = p / (kW4Out - kW4In);
    const int k  = p % (kW4Out - kW4In);
    out4 = bc * kRow4 + oh * kW4Out + kW4In + k;
  } else {
    // Bottom rows 256..303, full width
    const int q  = p - kPadRight;
    const int oh = kInterior + q / kW4Out;
    const int k  = q % kW4Out;
    out4 = bc * kRow4 + oh * kW4Out + k;
  }

  v4f z = {0.f, 0.f, 0.f, 0.f};
  __builtin_nontemporal_store(z, reinterpret_cast<v4f*>(out) + out4);
}

extern "C" void kernel_launch(void* const* d_in, const int* in_sizes, int n_in,
                              void* d_out, int out_size, void* d_ws,
                              size_t ws_size, hipStream_t stream) {
  (void)in_sizes; (void)n_in; (void)out_size; (void)d_ws; (void)ws_size;
  const float* x = reinterpret_cast<const float*>(d_in[0]);
  float* out     = reinterpret_cast<float*>(d_out);

  duc_interior_kernel<<<dim3(kInterior4 / kBlock), dim3(kBlock), 0, stream>>>(
      x, out);
  duc_pad_kernel<<<dim3(kPad4 / kBlock), dim3(kBlock), 0, stream>>>(out);
}